// RhoNN_44693429682830
// MI455X (gfx1250) — compile-verified
//
#include <hip/hip_runtime.h>

// CDNA5 / gfx1250, wave32. All matrix math via V_WMMA_F32_16X16X4_F32.
//
// Bit-exact simplification: softmax over a singleton axis == 1.0, so the
// reference's score path (W3/W4 GEMMs + per-edge dot products) is dead code.
//   out_a = w1a + 2*( segsum_{eaa}(w2a[j] + faa@W5aa + b5aa -> i)
//                   + segsum_{eoa}(w2o[j] + foa@W5oa + b5oa -> i) )
//   out_o = w1o
// followed by LayerNorm*g+b + residual, twice.
//
// LDS holds W pre-swizzled as K-pair-interleaved float2 so each lane's WMMA B
// fragment is a single aligned ds_load_b64 (no VALU repacking):
//   Wl2[(k>>1)*128 + col] = { W[k][col], W[k+1][col] }
// Lane (half = lane>>4) reads pair index t = 2*kb + half at col = nt*16+l16.

typedef float v2f __attribute__((ext_vector_type(2)));
typedef float v8f __attribute__((ext_vector_type(8)));

#define NAg   32768
#define HD    128
#define FEd   64
#define NEDGE 262144

// ---------------------------------------------------------------- zero
__global__ __launch_bounds__(256) void zero_f4(float4* __restrict__ p, int n4) {
  int t = blockIdx.x * blockDim.x + threadIdx.x;
  if (t < n4) { float4 z; z.x = 0.f; z.y = 0.f; z.z = 0.f; z.w = 0.f; p[t] = z; }
}

// stage W (rowsK x 128, row-major) into LDS as K-pair-interleaved float2
__device__ __forceinline__ void stage_w_swizzled(
    v2f* __restrict__ Wl2, const float* __restrict__ W, int kPairs, int tid)
{
  for (int idx = tid; idx < kPairs * (HD / 4); idx += 128) {
    const int t  = idx >> 5;                 // k-pair index
    const int c4 = idx & 31;                 // float4 column group
    const float4 r0 = ((const float4*)(W + (size_t)(2 * t    ) * HD))[c4];
    const float4 r1 = ((const float4*)(W + (size_t)(2 * t + 1) * HD))[c4];
    v2f* dst = &Wl2[t * HD + c4 * 4];
    v2f p0; p0.x = r0.x; p0.y = r1.x;
    v2f p1; p1.x = r0.y; p1.y = r1.y;
    v2f p2; p2.x = r0.z; p2.y = r1.z;
    v2f p3; p3.x = r0.w; p3.y = r1.w;
    dst[0] = p0; dst[1] = p1; dst[2] = p2; dst[3] = p3;
  }
}

// ---------------------------------------------------------------- Y = X(Mx128) @ W(128x128) + b
// block = 128 threads = 4 waves; wave -> 16 rows x 128 cols; block -> 64 rows.
__global__ __launch_bounds__(128) void gemm_node_128(
    const float* __restrict__ X, const float* __restrict__ W,
    const float* __restrict__ bias, float* __restrict__ Y)
{
  __shared__ v2f Wl[(HD / 2) * HD];          // 64 KB, swizzled
  const int tid = threadIdx.x;
  stage_w_swizzled(Wl, W, HD / 2, tid);
  __syncthreads();

  const int lane = tid & 31;
  const int wv   = tid >> 5;
  const int half = lane >> 4;                // 0|1
  const int l16  = lane & 15;
  const int kOff = half * 2;                 // A fragment K sub-offset (ISA layout)
  const int rowBase = blockIdx.x * 64 + wv * 16;

  v8f acc[8];
  #pragma unroll
  for (int nt = 0; nt < 8; ++nt) {
    const float bv = bias[nt * 16 + l16];
    #pragma unroll
    for (int r = 0; r < 8; ++r) acc[nt][r] = bv;
  }

  const float* Arow = X + (size_t)(rowBase + l16) * HD;
  const v2f*   Bbase = &Wl[half * HD + l16]; // + 2*HD per kb, + 16 per nt

  for (int kb = 0; kb < 32; ++kb) {          // K = 128, step 4
    v2f a = *(const v2f*)(Arow + kb * 4 + kOff);
    const v2f* bp = Bbase + 2 * HD * kb;
    #pragma unroll
    for (int nt = 0; nt < 8; ++nt) {
      v2f b = bp[nt * 16];                   // single ds_load_b64
      acc[nt] = __builtin_amdgcn_wmma_f32_16x16x4_f32(
          false, a, false, b, (short)0, acc[nt], false, false);
    }
  }

  // D layout: VGPR r, lane -> (M = r + 8*half, N = nt*16 + l16)
  #pragma unroll
  for (int nt = 0; nt < 8; ++nt) {
    const int col = nt * 16 + l16;
    #pragma unroll
    for (int r = 0; r < 8; ++r)
      Y[(size_t)(rowBase + r + 8 * half) * HD + col] = acc[nt][r];
  }
}

// ---------------------------------------------------------------- fused edge kernel
// Per wave: 16 edges. w5tile = feat[16x64] @ W5[64x128] + b5 (WMMA, W5 in LDS),
// then accOut[i[e]] += w5tile[e] + w2s[j[e]]  (fp32 HW atomics).
__global__ __launch_bounds__(128) void edge_w5_agg(
    const float* __restrict__ feat, const float* __restrict__ W5,
    const float* __restrict__ b5,   const int*   __restrict__ edges,
    const float* __restrict__ w2s,  float*       __restrict__ accOut)
{
  __shared__ v2f Wl[(FEd / 2) * HD];         // 32 KB, swizzled
  const int tid = threadIdx.x;
  stage_w_swizzled(Wl, W5, FEd / 2, tid);
  __syncthreads();

  const int lane = tid & 31;
  const int wv   = tid >> 5;
  const int half = lane >> 4;
  const int l16  = lane & 15;
  const int kOff = half * 2;
  const int eBase = blockIdx.x * 64 + wv * 16;

  v8f acc[8];
  #pragma unroll
  for (int nt = 0; nt < 8; ++nt) {
    const float bv = b5[nt * 16 + l16];
    #pragma unroll
    for (int r = 0; r < 8; ++r) acc[nt][r] = bv;
  }

  const float* Arow = feat + (size_t)(eBase + l16) * FEd;
  const v2f*   Bbase = &Wl[half * HD + l16];

  #pragma unroll
  for (int kb = 0; kb < 16; ++kb) {          // K = 64, step 4
    v2f a = *(const v2f*)(Arow + kb * 4 + kOff);
    const v2f* bp = Bbase + 2 * HD * kb;
    #pragma unroll
    for (int nt = 0; nt < 8; ++nt) {
      v2f b = bp[nt * 16];
      acc[nt] = __builtin_amdgcn_wmma_f32_16x16x4_f32(
          false, a, false, b, (short)0, acc[nt], false, false);
    }
  }

  // fuse gather(w2s[j]) + scatter-add(accOut[i]); raw sum, x2 applied at finalize
  #pragma unroll
  for (int r = 0; r < 8; ++r) {
    const int e  = eBase + r + 8 * half;     // edge index of this D row
    const int di = edges[2 * e];             // destination i
    const int sj = edges[2 * e + 1];         // source j
    const float* w2row = w2s    + (size_t)sj * HD;
    float*       orow  = accOut + (size_t)di * HD;
    #pragma unroll
    for (int nt = 0; nt < 8; ++nt) {
      const int col = nt * 16 + l16;
      const float v = acc[nt][r] + w2row[col];
      __hip_atomic_fetch_add(orow + col, v, __ATOMIC_RELAXED,
                             __HIP_MEMORY_SCOPE_AGENT);
    }
  }
}

// ---------------------------------------------------------------- out = LN(w1 [+ 2*acc])*g + b + xres
// one wave per 128-wide row; lane handles 4 cols (float4)
__global__ __launch_bounds__(128) void finalize_ln(
    const float* __restrict__ w1, const float* __restrict__ acc,
    const float* __restrict__ g,  const float* __restrict__ b,
    const float* __restrict__ xres, float* __restrict__ out, int hasAcc)
{
  const int lane = threadIdx.x & 31;
  const int row  = blockIdx.x * 4 + (threadIdx.x >> 5);

  float4 v = ((const float4*)(w1 + (size_t)row * HD))[lane];
  if (hasAcc) {
    float4 a = ((const float4*)(acc + (size_t)row * HD))[lane];
    v.x += 2.f * a.x; v.y += 2.f * a.y; v.z += 2.f * a.z; v.w += 2.f * a.w;
  }
  float s = v.x + v.y + v.z + v.w;
  for (int off = 16; off > 0; off >>= 1) s += __shfl_xor(s, off, 32);
  const float mu = s * (1.f / 128.f);
  const float dx = v.x - mu, dy = v.y - mu, dz = v.z - mu, dw = v.w - mu;
  float ss = dx * dx + dy * dy + dz * dz + dw * dw;
  for (int off = 16; off > 0; off >>= 1) ss += __shfl_xor(ss, off, 32);
  const float rs = rsqrtf(ss * (1.f / 128.f) + 1e-5f);

  const float4 gv = ((const float4*)g)[lane];
  const float4 bv = ((const float4*)b)[lane];
  const float4 xr = ((const float4*)(xres + (size_t)row * HD))[lane];
  float4 o;
  o.x = dx * rs * gv.x + bv.x + xr.x;
  o.y = dy * rs * gv.y + bv.y + xr.y;
  o.z = dz * rs * gv.z + bv.z + xr.z;
  o.w = dw * rs * gv.w + bv.w + xr.w;
  ((float4*)(out + (size_t)row * HD))[lane] = o;
}

// ---------------------------------------------------------------- host
// Input order: x_agent, x_object, feat_aa, feat_oa, feat_oo, edges_aa,
// edges_oa, edges_oo, then params in dict order (26 per layer, base 8 / 34):
//   +0 W1_A_w +1 W1_A_b +2 W1_O_w +3 W1_O_b +4 W2_A_w +5 W2_A_b +6 W2_O_w
//   +7 W2_O_b +8..15 W3/W4 (dead) +16 W5_AA_w +17 W5_AA_b +18 W5_OA_w
//   +19 W5_OA_b +20/21 W5_OO (dead) +22 ln_A_g +23 ln_A_b +24 ln_O_g +25 ln_O_b
extern "C" void kernel_launch(void* const* d_in, const int* in_sizes, int n_in,
                              void* d_out, int out_size, void* d_ws, size_t ws_size,
                              hipStream_t stream)
{
  (void)in_sizes; (void)n_in; (void)out_size; (void)ws_size;
  const float* x_agent  = (const float*)d_in[0];
  const float* x_object = (const float*)d_in[1];
  const float* feat_aa  = (const float*)d_in[2];
  const float* feat_oa  = (const float*)d_in[3];
  const int*   edges_aa = (const int*)d_in[5];
  const int*   edges_oa = (const int*)d_in[6];
  auto P = [&](int i) { return (const float*)d_in[i]; };

  const size_t NM = (size_t)NAg * HD;        // 4,194,304 floats
  float* ws  = (float*)d_ws;                 // needs 7*NM*4 = 112 MB
  float* acc = ws;
  float* w1a = ws + 1 * NM;
  float* w2a = ws + 2 * NM;
  float* w1o = ws + 3 * NM;
  float* w2o = ws + 4 * NM;
  float* z1a = ws + 5 * NM;
  float* z1o = ws + 6 * NM;
  float* out_a = (float*)d_out;
  float* out_o = (float*)d_out + NM;

  struct LayerIO { const float* xa; const float* xo; float* za; float* zo; int B; };
  const LayerIO L[2] = { { x_agent, x_object, z1a,   z1o,   8  },
                         { z1a,     z1o,      out_a, out_o, 34 } };

  for (int li = 0; li < 2; ++li) {
    const int B = L[li].B;
    zero_f4<<<(int)(NM / 4 / 256), 256, 0, stream>>>((float4*)acc, (int)(NM / 4));
    gemm_node_128<<<NAg / 64, 128, 0, stream>>>(L[li].xa, P(B + 0), P(B + 1), w1a);
    gemm_node_128<<<NAg / 64, 128, 0, stream>>>(L[li].xa, P(B + 4), P(B + 5), w2a);
    gemm_node_128<<<NAg / 64, 128, 0, stream>>>(L[li].xo, P(B + 2), P(B + 3), w1o);
    gemm_node_128<<<NAg / 64, 128, 0, stream>>>(L[li].xo, P(B + 6), P(B + 7), w2o);
    edge_w5_agg<<<NEDGE / 64, 128, 0, stream>>>(feat_aa, P(B + 16), P(B + 17),
                                                edges_aa, w2a, acc);
    edge_w5_agg<<<NEDGE / 64, 128, 0, stream>>>(feat_oa, P(B + 18), P(B + 19),
                                                edges_oa, w2o, acc);
    finalize_ln<<<NAg / 4, 128, 0, stream>>>(w1a, acc, P(B + 22), P(B + 23),
                                             L[li].xa, L[li].za, 1);
    finalize_ln<<<NAg / 4, 128, 0, stream>>>(w1o, nullptr, P(B + 24), P(B + 25),
                                             L[li].xo, L[li].zo, 0);
  }
}